// Convolution_42159398977944
// MI455X (gfx1250) — compile-verified
//
#include <hip/hip_runtime.h>

// CDNA5 / gfx1250, wave32. FP32 GEMM-conv via V_WMMA_F32_16X16X4_F32.
// Each wave: 32(M) x 32(N) output tile = 4 accumulators -> 4 WMMAs per
// k-step against 2 global b64 (A) + 2 ds b64 (B) loads.

typedef __attribute__((ext_vector_type(2))) float v2f;
typedef __attribute__((ext_vector_type(8))) float v8f;

#define SEQ    200000
#define EMB    300
#define WIN    3
#define KTOT   (WIN * EMB)          // 900
#define NF     32
#define NOUT   (SEQ - WIN + 1)      // 199998
#define KSTEPS (KTOT / 4)           // 225 k-steps of 4
#define WLDS_FLOATS (KTOT * NF)     // 28800 floats = 115200 bytes

__global__ __launch_bounds__(128)
void conv_im2col_wmma_f32(const float* __restrict__ in,   // [SEQ, EMB] row-major
                          const float* __restrict__ W,    // [NF, KTOT] row-major
                          const float* __restrict__ bias, // [NF]
                          float* __restrict__ out)        // [NF, NOUT]
{
    extern __shared__ float wlds[]; // W transposed: (k,n) -> ((k>>1)*NF + n)*2 + (k&1)

    const int tid  = threadIdx.x;
    const int lane = tid & 31;
    const int nlo  = lane & 15;   // N index (low tile) / A-row selector
    const int kh   = lane >> 4;   // 0: K pair {k0,k0+1}, 1: K pair {k0+2,k0+3}
    const int wave = tid >> 5;

    // ---- Stage W into LDS, transposed into b64-friendly pairs (coalesced read) ----
    for (int idx = tid; idx < WLDS_FLOATS; idx += 128) {
        int f = idx / KTOT;          // filter (row of W)
        int k = idx - f * KTOT;      // column of W = K index
        wlds[((k >> 1) * NF + f) * 2 + (k & 1)] = W[idx];
    }
    __syncthreads();

    // ---- Per-wave 32(M) x 32(N) tile: two 16-row M-tiles x two 16-col N-tiles ----
    const int m0 = (blockIdx.x * 4 + wave) * 32;

    // A layout (32-bit A 16x4): lanes 0-15 & 16-31 both hold M = base + (lane&15);
    // VGPR0/1 hold K = k0+2*kh, k0+2*kh+1. Clamp rows so EXEC stays all-ones.
    int mA0 = m0 + nlo;
    int mA1 = m0 + 16 + nlo;
    if (mA0 > NOUT - 1) mA0 = NOUT - 1;
    if (mA1 > NOUT - 1) mA1 = NOUT - 1;
    const float* aPtr0 = in + mA0 * EMB + 2 * kh;
    const float* aPtr1 = in + mA1 * EMB + 2 * kh;

    // B fragment base: pair index (2*kk + kh), column nlo (tile0) / nlo+16 (tile1)
    const int base0 = (kh * NF + nlo) * 2;

    v8f c00 = {};   // M-tile0 x N 0..15
    v8f c01 = {};   // M-tile0 x N 16..31
    v8f c10 = {};   // M-tile1 x N 0..15
    v8f c11 = {};   // M-tile1 x N 16..31

#pragma unroll 5
    for (int kk = 0; kk < KSTEPS; ++kk) {
        v2f a0 = *(const v2f*)(aPtr0 + 4 * kk);                      // global_load_b64
        v2f a1 = *(const v2f*)(aPtr1 + 4 * kk);                      // global_load_b64
        v2f b0 = *(const v2f*)(wlds + base0 + 128 * kk);             // ds_load_b64
        v2f b1 = *(const v2f*)(wlds + base0 + 32 + 128 * kk);        // ds_load_b64
        // (neg_a, A, neg_b, B, c_mod, C, reuse_a, reuse_b)
        c00 = __builtin_amdgcn_wmma_f32_16x16x4_f32(false, a0, false, b0,
                                                    (short)0, c00, false, false);
        c01 = __builtin_amdgcn_wmma_f32_16x16x4_f32(false, a0, false, b1,
                                                    (short)0, c01, false, false);
        c10 = __builtin_amdgcn_wmma_f32_16x16x4_f32(false, a1, false, b0,
                                                    (short)0, c10, false, false);
        c11 = __builtin_amdgcn_wmma_f32_16x16x4_f32(false, a1, false, b1,
                                                    (short)0, c11, false, false);
    }

    // ---- Epilogue: bias + ReLU, store transposed out[n, m] (nontemporal) ----
    const float bz0 = bias[nlo];
    const float bz1 = bias[nlo + 16];

#pragma unroll
    for (int v = 0; v < 8; ++v) {
        // D layout: VGPR v -> M = base + v + 8*kh, N = lane&15 (per N-tile)
        int ma = m0 + v + 8 * kh;
        int mb = m0 + 16 + v + 8 * kh;
        if (ma < NOUT) {
            float r0 = c00[v] + bz0; r0 = r0 > 0.0f ? r0 : 0.0f;
            float r1 = c01[v] + bz1; r1 = r1 > 0.0f ? r1 : 0.0f;
            __builtin_nontemporal_store(r0, out + nlo * NOUT + ma);
            __builtin_nontemporal_store(r1, out + (nlo + 16) * NOUT + ma);
        }
        if (mb < NOUT) {
            float r0 = c10[v] + bz0; r0 = r0 > 0.0f ? r0 : 0.0f;
            float r1 = c11[v] + bz1; r1 = r1 > 0.0f ? r1 : 0.0f;
            __builtin_nontemporal_store(r0, out + nlo * NOUT + mb);
            __builtin_nontemporal_store(r1, out + (nlo + 16) * NOUT + mb);
        }
    }
}

extern "C" void kernel_launch(void* const* d_in, const int* in_sizes, int n_in,
                              void* d_out, int out_size, void* d_ws, size_t ws_size,
                              hipStream_t stream) {
    (void)in_sizes; (void)n_in; (void)out_size; (void)d_ws; (void)ws_size;

    const float* in_seq = (const float*)d_in[0]; // [SEQ*EMB]
    const float* W      = (const float*)d_in[1]; // [NF*KTOT]
    const float* b      = (const float*)d_in[2]; // [NF]
    float*       out    = (float*)d_out;         // [NF*NOUT]

    // 128 M-rows per block (4 waves x 32 rows)
    const int blocks = (NOUT + 127) / 128;   // 1563

    conv_im2col_wmma_f32<<<blocks, 128, WLDS_FLOATS * sizeof(float), stream>>>(
        in_seq, W, b, out);
}